// MeshShuffle_SSR2_76819785056407
// MI455X (gfx1250) — compile-verified
//
#include <hip/hip_runtime.h>
#include <hip/hip_bf16.h>
#include <stdint.h>

// Problem constants (from reference)
#define BB    4
#define NCH   32          // output channels
#define NV    163842      // vertices
#define EE    163840
#define K3    (3 * EE)    // 491520 gathered outputs per (b,c)
#define OUTW  (NV + K3)   // 655362

#define K_TILE 256        // gather tile (k's per block)
#define V_TILE 64         // pass1 vertex tile

typedef uint32_t u32x4 __attribute__((ext_vector_type(4)));
typedef uint32_t u32x8 __attribute__((ext_vector_type(8)));

// ---------------------------------------------------------------------------
// Pass 1: out[:, :, 0:NV] = 0.5*(x[0:32]+x[32:64]);
//         inpT[b][v][c]   = 0.25*(x[64+c]+x[96+c])   (transposed layout)
// x is streamed once -> NT loads; out is write-once -> NT stores;
// inpT keeps default RT so the 84 MB gather target stays L2-resident.
// ---------------------------------------------------------------------------
__global__ void pass1_part1_transpose(const float* __restrict__ x,
                                      float* __restrict__ out,
                                      float* __restrict__ inpT) {
    __shared__ float tile[V_TILE * 33];     // [v_local][c], pad 33 -> no bank conflicts
    const int b   = blockIdx.y;
    const int v0  = blockIdx.x * V_TILE;
    const int tid = threadIdx.x;

    // load/compute phase: lanes along v (coalesced 128B), 4 channel groups x 8
    const int vl = tid & (V_TILE - 1);
    const int cg = tid >> 6;                 // 0..3
    const int v  = v0 + vl;
    if (v < NV) {
        const float* xb = x + (size_t)b * 128 * NV;
        float* ob = out + (size_t)b * NCH * OUTW;
#pragma unroll
        for (int i = 0; i < 8; ++i) {
            const int c = cg * 8 + i;                       // 0..31
            const float a0 = __builtin_nontemporal_load(&xb[(size_t)c        * NV + v]);
            const float a1 = __builtin_nontemporal_load(&xb[(size_t)(c + 32) * NV + v]);
            __builtin_nontemporal_store(0.5f * (a0 + a1),
                                        &ob[(size_t)c * OUTW + v]);   // part1, coalesced
            const float b0 = __builtin_nontemporal_load(&xb[(size_t)(c + 64) * NV + v]);
            const float b1 = __builtin_nontemporal_load(&xb[(size_t)(c + 96) * NV + v]);
            tile[vl * 33 + c] = 0.25f * (b0 + b1);          // inp, staged for transpose
        }
    }
    __syncthreads();

    // write phase: inpT is contiguous over (v,c) -> 1KB fully coalesced RT stores
    const int c2 = tid & 31;
    const int vo = tid >> 5;                 // 0..7
#pragma unroll
    for (int i = 0; i < 8; ++i) {
        const int vl2 = vo + i * 8;
        const int vv  = v0 + vl2;
        if (vv < NV)
            inpT[((size_t)b * NV + vv) * NCH + c2] = tile[vl2 * 33 + c2];
    }
}

// ---------------------------------------------------------------------------
// Pass 2: idxPairs[k] = ( ssi[2*unique[k]], ssi[2*unique[k]+1] ) as int32
// (the (3,E) reshape/flatten in the reference cancels: flat(v0)[u] == ssi[2u])
// ---------------------------------------------------------------------------
__global__ void pass2_indices(const long long* __restrict__ ssi,
                              const long long* __restrict__ unique,
                              int2* __restrict__ idxPairs) {
    const int k = blockIdx.x * blockDim.x + threadIdx.x;
    if (k >= K3) return;
    const long long u = __builtin_nontemporal_load(&unique[k]);  // read-once stream
    const long long* p = ssi + (size_t)u * 2;                    // gathered, L2-resident
    idxPairs[k] = make_int2((int)p[0], (int)p[1]);
}

// ---------------------------------------------------------------------------
// Pass 3: out[b, c, NV + k] = inpT[b][v0][c] + inpT[b][v1][c]
// Index tile staged into LDS with the Tensor Data Mover (async-tensor path);
// lane == channel (wave32), gathers are 128B-coalesced lines from L2-resident
// inpT, results transposed back through padded LDS so the global NT stores
// along k are coalesced.
// ---------------------------------------------------------------------------
struct Smem3 {
    int2  idx[K_TILE];            // at LDS offset 0 (TDM destination)
    float data[K_TILE * 33];      // transpose staging, pad 33
};

__global__ void pass3_gather(const float* __restrict__ inpT,
                             const int2* __restrict__ idxPairs,
                             float* __restrict__ out) {
    __shared__ Smem3 smem;
    const int b         = blockIdx.y;
    const int tileStart = blockIdx.x * K_TILE;
    const int tid       = threadIdx.x;

    if (tid < 32) {
        // Build Tensor DMA Descriptor (D#) in SGPRs: 1 row of K_TILE 8-byte
        // elements, global -> LDS offset 0. workgroup_mask=0 (not in cluster).
        const uint64_t gaddr = (uint64_t)(const void*)(idxPairs + tileStart);
        u32x4 g0;
        g0[0] = 1u;                                           // count=1
        g0[1] = 0u;                                           // lds_addr = 0 (smem.idx)
        g0[2] = (uint32_t)gaddr;                              // global_addr[31:0]
        g0[3] = ((uint32_t)(gaddr >> 32) & 0x1FFFFFFu)        // global_addr[56:32]
                | (2u << 30);                                 // type=2 (image)
        u32x8 g1;
        g1[0] = 3u << 16;                                     // data_size=3 (8B)
        g1[1] = (uint32_t)K_TILE << 16;                       // tensor_dim0 = K_TILE
        g1[2] = 1u << 16;                                     // tensor_dim1 = 1
        g1[3] = (uint32_t)K_TILE << 16;                       // tile_dim0 = K_TILE
        g1[4] = 1u | (1u << 16);                              // tile_dim1=1, tile_dim2=1
        g1[5] = (uint32_t)K_TILE;                             // dim0_stride = K_TILE
        g1[6] = (uint32_t)K_TILE << 16;                       // dim1_stride = K_TILE
        g1[7] = 0u;
        asm volatile("tensor_load_to_lds %0, %1" :: "s"(g0), "s"(g1) : "memory");
        __builtin_amdgcn_s_wait_tensorcnt(0);
    }
    __syncthreads();

    // gather phase: each wave owns one k per iteration, lane == channel.
    const int lane = tid & 31;
    const int wave = tid >> 5;                // 0..7
    const float* basep = inpT + (size_t)b * NV * NCH;
#pragma unroll 4
    for (int kl = wave; kl < K_TILE; kl += 8) {
        const int2 vv = smem.idx[kl];         // LDS broadcast read
        const float val = basep[(size_t)vv.x * NCH + lane]    // 128B coalesced line
                        + basep[(size_t)vv.y * NCH + lane];   // 128B coalesced line
        smem.data[kl * 33 + lane] = val;      // conflict-free
    }
    __syncthreads();

    // write phase: k on lanes -> 1KB coalesced NT global stores per channel row
    float* ob = out + (size_t)b * NCH * OUTW + NV + tileStart;
#pragma unroll 4
    for (int c = 0; c < NCH; ++c) {
        __builtin_nontemporal_store(smem.data[tid * 33 + c],  // stride-33: conflict-free
                                    &ob[(size_t)c * OUTW + tid]);
    }
}

// ---------------------------------------------------------------------------
extern "C" void kernel_launch(void* const* d_in, const int* in_sizes, int n_in,
                              void* d_out, int out_size, void* d_ws, size_t ws_size,
                              hipStream_t stream) {
    (void)in_sizes; (void)n_in; (void)out_size; (void)ws_size;
    const float*     x      = (const float*)d_in[0];
    const long long* ssi    = (const long long*)d_in[1];   // (3,E,2) int64
    const long long* unique = (const long long*)d_in[2];   // (3E,)   int64
    float* out = (float*)d_out;

    // workspace: inpT (B*NV*32 floats, 83.9 MB) then idxPairs (K3 int2, 3.9 MB)
    float* inpT = (float*)d_ws;
    const size_t inpT_bytes = (size_t)BB * NV * NCH * sizeof(float); // 256B aligned
    int2* idxPairs = (int2*)((char*)d_ws + inpT_bytes);

    // pass 1: part1 + transposed inp
    {
        dim3 grid((NV + V_TILE - 1) / V_TILE, BB);   // (2561, 4)
        pass1_part1_transpose<<<grid, 256, 0, stream>>>(x, out, inpT);
    }
    // pass 2: fused gather indices
    {
        dim3 grid((K3 + 255) / 256);                 // 1920
        pass2_indices<<<grid, 256, 0, stream>>>(ssi, unique, idxPairs);
    }
    // pass 3: TDM-staged gather
    {
        dim3 grid(K3 / K_TILE, BB);                  // (1920, 4)
        pass3_gather<<<grid, 256, 0, stream>>>(inpT, idxPairs, out);
    }
}